// SinkhornSolver_40973988004393
// MI455X (gfx1250) — compile-verified
//
#include <hip/hip_runtime.h>
#include <hip/hip_bf16.h>
#include <math.h>

typedef __attribute__((ext_vector_type(2))) float     v2f;
typedef __attribute__((ext_vector_type(8))) float     v8f;
typedef __attribute__((ext_vector_type(4))) float     f4v;
typedef __attribute__((ext_vector_type(2))) _Float16  h2v;
typedef __attribute__((ext_vector_type(8))) _Float16  h8v;

#define NMAT      8192
#define DIMK      128
#define EPSW      10.0f
#define INV_EPS   0.1f
#define TOLW      1e-6f
#define NITERS    10
#define ROW_CHUNKS 64           // colsum row-chunking: 64 * 128 = 8192
#define ROWS_PER_CHUNK 128

// ---------------------------------------------------------------------------
// init: u = v = 1, done flag = 0
// ---------------------------------------------------------------------------
__global__ void k_init_vec(float* __restrict__ u, float* __restrict__ v,
                           int* __restrict__ flags) {
    int i = blockIdx.x * 256 + threadIdx.x;
    u[i] = 1.0f;
    v[i] = 1.0f;
    if (i == 0) { flags[0] = 0; flags[1] = 0; }
}

// ---------------------------------------------------------------------------
// squared row norms of x and y
// ---------------------------------------------------------------------------
__global__ void k_sqnorm(const float* __restrict__ x, const float* __restrict__ y,
                         float* __restrict__ x2, float* __restrict__ y2) {
    int t = blockIdx.x * 256 + threadIdx.x;
    const float* p = (t < NMAT) ? (x + (size_t)t * DIMK)
                                : (y + (size_t)(t - NMAT) * DIMK);
    float s = 0.0f;
    #pragma unroll 4
    for (int k = 0; k < DIMK; k += 4) {
        f4v q = *(const f4v*)(p + k);
        s += q.x * q.x + q.y * q.y + q.z * q.z + q.w * q.w;
    }
    if (t < NMAT) x2[t] = s; else y2[t - NMAT] = s;
}

// ---------------------------------------------------------------------------
// WMMA fp32 GEMM: G = x @ y^T, tile 16x16 per wave, K=128 via 32x 16x16x4.
// Epilogue: d2 = x2 + y2 - 2G ; C = sqrt(max(d2,0)) ; K = exp(-C/eps) -> f16.
// A-layout (16x4 f32): lanes0-15 M=0..15 {K0,K1}, lanes16-31 {K2,K3}.
// ---------------------------------------------------------------------------
__global__ void k_gemm_expK(const float* __restrict__ x, const float* __restrict__ y,
                            const float* __restrict__ x2, const float* __restrict__ y2,
                            _Float16* __restrict__ Kh) {
    int wave = threadIdx.x >> 5;
    int lane = threadIdx.x & 31;
    int tile = blockIdx.x * 8 + wave;          // 512*512 tiles total
    int ti = tile >> 9;                        // tile row
    int tj = tile & 511;                       // tile col
    int r  = lane & 15;
    int kh = lane >> 4;                        // 0: K pair {0,1}; 1: K pair {2,3}

    const float* xr = x + (size_t)(ti * 16 + r) * DIMK + kh * 2;
    const float* yr = y + (size_t)(tj * 16 + r) * DIMK + kh * 2;

    v8f acc = {};
    #pragma unroll 4
    for (int k0 = 0; k0 < DIMK; k0 += 4) {
        v2f a = *(const v2f*)(xr + k0);
        v2f b = *(const v2f*)(yr + k0);
        acc = __builtin_amdgcn_wmma_f32_16x16x4_f32(
            /*neg_a=*/false, a, /*neg_b=*/false, b,
            /*c_mod=*/(short)0, acc, /*reuse_a=*/false, /*reuse_b=*/false);
    }

    // C/D layout: VGPR m, lanes0-15 -> M=m, lanes16-31 -> M=m+8; N=lane&15
    int col = tj * 16 + r;
    float y2c = y2[col];
    int rowBase = ti * 16 + kh * 8;
    #pragma unroll
    for (int m = 0; m < 8; ++m) {
        int row = rowBase + m;
        float d2 = x2[row] + y2c - 2.0f * acc[m];
        float c  = sqrtf(fmaxf(d2, 0.0f));
        Kh[(size_t)row * NMAT + col] = (_Float16)__expf(-INV_EPS * c);
    }
}

// ---------------------------------------------------------------------------
// u_new = a / (K @ v)   (one wave per row; v staged in LDS)
// also emits per-block partial of ||u_new - u||^2
// ---------------------------------------------------------------------------
__global__ void k_rowsum(const _Float16* __restrict__ Kh,
                         const float* __restrict__ v,
                         const float* __restrict__ u,
                         float* __restrict__ un,
                         float* __restrict__ normPart) {
    __shared__ float vsh[NMAT];
    __shared__ float wsum[8];
    int tid = threadIdx.x;
    for (int j = tid; j < NMAT; j += 256) vsh[j] = v[j];
    __syncthreads();

    int wave = tid >> 5, lane = tid & 31;
    int row = blockIdx.x * 8 + wave;
    const _Float16* kr = Kh + (size_t)row * NMAT;

    float acc = 0.0f;
    for (int j0 = 0; j0 < NMAT; j0 += 256) {
        int jb = j0 + lane * 8;
        __builtin_prefetch(kr + jb + 1024, 0, 0);     // global_prefetch_b8
        h8v kv = *(const h8v*)(kr + jb);
        #pragma unroll
        for (int t = 0; t < 8; ++t) acc += (float)kv[t] * vsh[jb + t];
    }
    #pragma unroll
    for (int off = 16; off > 0; off >>= 1) acc += __shfl_down(acc, off, 32);

    if (lane == 0) {
        float unv = (1.0f / (float)NMAT) / acc;
        un[row] = unv;
        float d = unv - u[row];
        wsum[wave] = d * d;
    }
    __syncthreads();
    if (tid == 0) {
        float s = 0.0f;
        #pragma unroll
        for (int w = 0; w < 8; ++w) s += wsum[w];
        normPart[blockIdx.x] = s;
    }
}

// ---------------------------------------------------------------------------
// convergence bookkeeping (torch semantics: update with OLD done, then OR)
// flags[0] = done ; flags[1] = commit-this-iteration (= !old_done)
// ---------------------------------------------------------------------------
__global__ void k_norm_done(const float* __restrict__ normPart, int* __restrict__ flags) {
    __shared__ float sh[256];
    int tid = threadIdx.x;
    float s = 0.0f;
    for (int i = tid; i < 1024; i += 256) s += normPart[i];
    sh[tid] = s; __syncthreads();
    for (int off = 128; off > 0; off >>= 1) {
        if (tid < off) sh[tid] += sh[tid + off];
        __syncthreads();
    }
    if (tid == 0) {
        int done = flags[0];
        int conv = (sqrtf(sh[0]) < TOLW) ? 1 : 0;
        flags[1] = done ? 0 : 1;
        flags[0] = done | conv;
    }
}

// ---------------------------------------------------------------------------
// partial column sums of K^T @ u_new  (grid: 16 col-groups x 64 row-chunks)
// ---------------------------------------------------------------------------
__global__ void k_colsum(const _Float16* __restrict__ Kh,
                         const float* __restrict__ un,
                         float* __restrict__ colPart) {
    int j  = blockIdx.x * 512 + threadIdx.x * 2;
    int i0 = blockIdx.y * ROWS_PER_CHUNK;
    float ax = 0.0f, ay = 0.0f;
    #pragma unroll 4
    for (int i = i0; i < i0 + ROWS_PER_CHUNK; ++i) {
        float uu = un[i];
        h2v kv = *(const h2v*)(Kh + (size_t)i * NMAT + j);
        ax += (float)kv.x * uu;
        ay += (float)kv.y * uu;
    }
    colPart[(size_t)blockIdx.y * NMAT + j]     = ax;
    colPart[(size_t)blockIdx.y * NMAT + j + 1] = ay;
}

// ---------------------------------------------------------------------------
// v_new = b / sum_chunks ; conditional commit of (u,v) per freeze flag
// ---------------------------------------------------------------------------
__global__ void k_vnew_commit(const float* __restrict__ colPart,
                              const float* __restrict__ un,
                              float* __restrict__ u, float* __restrict__ v,
                              const int* __restrict__ flags) {
    int j = blockIdx.x * 256 + threadIdx.x;
    float s = 0.0f;
    #pragma unroll 8
    for (int c = 0; c < ROW_CHUNKS; ++c) s += colPart[(size_t)c * NMAT + j];
    float vv = (1.0f / (float)NMAT) / s;
    if (flags[1]) { v[j] = vv; u[j] = un[j]; }
}

// ---------------------------------------------------------------------------
// P = u K v ; C = -eps*ln(K) ; partials of transport cost and sum(P*ln(P+1e-9))
// ---------------------------------------------------------------------------
__global__ void k_final(const _Float16* __restrict__ Kh,
                        const float* __restrict__ u, const float* __restrict__ v,
                        float* __restrict__ out,
                        float* __restrict__ tcPart, float* __restrict__ entPart) {
    __shared__ float shtc[256], shent[256];
    int tid = threadIdx.x;
    float tc = 0.0f, ent = 0.0f;
    for (int r = 0; r < 8; ++r) {
        int row = blockIdx.x * 8 + r;
        float ur = u[row];
        const _Float16* kr = Kh + (size_t)row * NMAT;
        float* pr = out + 1 + (size_t)row * NMAT;
        for (int j0 = 0; j0 < NMAT; j0 += 512) {
            int j = j0 + tid * 2;
            h2v kv = *(const h2v*)(kr + j);
            float k0 = (float)kv.x, k1 = (float)kv.y;
            float c0 = -EPSW * __logf(k0);
            float c1 = -EPSW * __logf(k1);
            float p0 = ur * k0 * v[j];
            float p1 = ur * k1 * v[j + 1];
            *(float2*)(pr + j) = make_float2(p0, p1);
            tc  += p0 * c0 + p1 * c1;
            ent += p0 * __logf(p0 + 1e-9f) + p1 * __logf(p1 + 1e-9f);
        }
    }
    shtc[tid] = tc; shent[tid] = ent; __syncthreads();
    for (int off = 128; off > 0; off >>= 1) {
        if (tid < off) { shtc[tid] += shtc[tid + off]; shent[tid] += shent[tid + off]; }
        __syncthreads();
    }
    if (tid == 0) { tcPart[blockIdx.x] = shtc[0]; entPart[blockIdx.x] = shent[0]; }
}

__global__ void k_reduce(const float* __restrict__ tcPart,
                         const float* __restrict__ entPart,
                         float* __restrict__ out) {
    __shared__ float s1[256], s2[256];
    int tid = threadIdx.x;
    float a = 0.0f, b = 0.0f;
    for (int i = tid; i < 1024; i += 256) { a += tcPart[i]; b += entPart[i]; }
    s1[tid] = a; s2[tid] = b; __syncthreads();
    for (int off = 128; off > 0; off >>= 1) {
        if (tid < off) { s1[tid] += s1[tid + off]; s2[tid] += s2[tid + off]; }
        __syncthreads();
    }
    // divergence = transport + EPS * (-sum P log(P+1e-9))
    if (tid == 0) out[0] = s1[0] - EPSW * s2[0];
}

// ---------------------------------------------------------------------------
extern "C" void kernel_launch(void* const* d_in, const int* in_sizes, int n_in,
                              void* d_out, int out_size, void* d_ws, size_t ws_size,
                              hipStream_t stream) {
    const float* x = (const float*)d_in[0];
    const float* y = (const float*)d_in[1];
    float* out = (float*)d_out;

    char* ws = (char*)d_ws;
    size_t off = 0;
    _Float16* Kh = (_Float16*)(ws + off);   off += (size_t)NMAT * NMAT * sizeof(_Float16); // 128 MB
    float* x2       = (float*)(ws + off);   off += NMAT * sizeof(float);
    float* y2       = (float*)(ws + off);   off += NMAT * sizeof(float);
    float* u        = (float*)(ws + off);   off += NMAT * sizeof(float);
    float* v        = (float*)(ws + off);   off += NMAT * sizeof(float);
    float* un       = (float*)(ws + off);   off += NMAT * sizeof(float);
    float* colPart  = (float*)(ws + off);   off += (size_t)ROW_CHUNKS * NMAT * sizeof(float);
    float* normPart = (float*)(ws + off);   off += 1024 * sizeof(float);
    float* tcPart   = (float*)(ws + off);   off += 1024 * sizeof(float);
    float* entPart  = (float*)(ws + off);   off += 1024 * sizeof(float);
    int*   flags    = (int*)(ws + off);     off += 2 * sizeof(int);

    k_init_vec<<<32, 256, 0, stream>>>(u, v, flags);
    k_sqnorm<<<64, 256, 0, stream>>>(x, y, x2, y2);
    k_gemm_expK<<<(512 * 512) / 8, 256, 0, stream>>>(x, y, x2, y2, Kh);

    for (int it = 0; it < NITERS; ++it) {
        k_rowsum<<<1024, 256, 0, stream>>>(Kh, v, u, un, normPart);
        k_norm_done<<<1, 256, 0, stream>>>(normPart, flags);
        k_colsum<<<dim3(16, 64), 256, 0, stream>>>(Kh, un, colPart);
        k_vnew_commit<<<32, 256, 0, stream>>>(colPart, un, u, v, flags);
    }

    k_final<<<1024, 256, 0, stream>>>(Kh, u, v, out, tcPart, entPart);
    k_reduce<<<1, 256, 0, stream>>>(tcPart, entPart, out);
}